// DEFORM_func_55353538511634
// MI455X (gfx1250) — compile-verified
//
#include <hip/hip_runtime.h>
#include <math.h>

// Problem shape (fixed by the reference's setup_inputs)
#define B_RODS 32768
#define NV     128
#define NE     127

#define RPB    256              // rods (threads) per block
#define CH     8                // vertices per LDS chunk
#define NCH    (NV / CH)        // 16 chunks
#define ROWF   28               // floats per rod-chunk row in LDS: 24 data + 4 pad (112B, 16B aligned)
#define SEGS   6                // 16B segments per rod-chunk (8 verts * 12B = 96B)

#define STIFF_THR 1e-10f
#define EPSN      1e-12f
#define ERRC      1e-6f

// ---- gfx1250 async global->LDS path (ASYNCcnt), with compile-safe fallback ----
#if defined(__HIP_DEVICE_COMPILE__) && defined(__gfx1250__) && \
    __has_builtin(__builtin_amdgcn_global_load_async_to_lds_b128) && \
    __has_builtin(__builtin_amdgcn_s_wait_asynccnt)
#  define USE_ASYNC 1
#else
#  define USE_ASYNC 0
#  if defined(__HIP_DEVICE_COMPILE__)
#    warning "gfx1250 async global->LDS builtins unavailable in device pass; synchronous LDS staging fallback in use"
#  endif
#endif

// Pointer types matching the builtin's prototype:
//   param0: int __vector(4) addrspace(1)*  (global source)
//   param1: LDS destination (AS3)
typedef int v4i __attribute__((vector_size(16)));
typedef __attribute__((address_space(1))) v4i* gas_v4i_ptr;
typedef __attribute__((address_space(3))) v4i* las_v4i_ptr;

__device__ __forceinline__ void cross3(float ax, float ay, float az,
                                       float bx, float by, float bz,
                                       float& cx, float& cy, float& cz) {
  cx = ay * bz - az * by;
  cy = az * bx - ax * bz;
  cz = ax * by - ay * bx;
}

__global__ __launch_bounds__(RPB) void der_rod_kernel(
    const float* __restrict__ vert,        // (B, V, 3)
    const float* __restrict__ initd,       // (B, 3)
    const float* __restrict__ restEdgeL,   // (B, E)
    const float* __restrict__ restRegionL, // (B, E)
    const float* __restrict__ wprev,       // (B, E, 2)
    const float* __restrict__ wnext,       // (B, E, 2)
    const float* __restrict__ theta,       // (B, E)
    const float* __restrict__ bendS,       // (1, E)
    const float* __restrict__ twistS,      // (1, E)
    float* __restrict__ om1,               // (B, E, 3)
    float* __restrict__ om2,               // (B, E, 3)
    float* __restrict__ okb,               // (B, E, 3)
    float* __restrict__ oen)               // (B, 1)
{
  __shared__ float sv[2][RPB * ROWF];      // 2 * 256 * 28 * 4 = 57344 B

  const int tid = threadIdx.x;
  const int rodBase = blockIdx.x * RPB;
  const int b = rodBase + tid;

  // Cooperative transposing copy: chunk c of all 256 rods of this block -> LDS buf.
  // Each thread moves 6 x 16B segments; consecutive lanes touch consecutive segments.
  auto issue = [&](int c, int buf) {
#pragma unroll
    for (int k = 0; k < SEGS; ++k) {
      const int s = tid + k * RPB;           // 0 .. 1535
      const int q = s / SEGS;                // block-local rod 0..255
      const int off = s - q * SEGS;          // 16B segment within rod-chunk
      const float* src =
          vert + ((size_t)(rodBase + q) * NV + (size_t)c * CH) * 3 + off * 4;
      float* dst = &sv[buf][q * ROWF + off * 4];
#if USE_ASYNC
      __builtin_amdgcn_global_load_async_to_lds_b128(
          (gas_v4i_ptr)src, (las_v4i_ptr)dst, 0, 0);
#else
      *(float4*)dst = *(const float4*)src;
#endif
    }
  };

  // Rod-constant data
  const float dx = initd[b * 3 + 0];
  const float dy = initd[b * 3 + 1];
  const float dz = initd[b * 3 + 2];
  const float bs_last = fmaxf(bendS[NE - 1], STIFF_THR);
  const size_t ebase = (size_t)b * NE;

  issue(0, 0);

  float ux = 0.f, uy = 0.f, uz = 0.f;      // bishop u
  float vx = 0.f, vy = 0.f, vz = 0.f;      // bishop v
  float pex = 0.f, pey = 0.f, pez = 0.f;   // previous edge
  float pvx = 0.f, pvy = 0.f, pvz = 0.f;   // previous vertex
  float prl = 0.f, pth = 0.f;              // previous restEdgeL / theta
  float energy = 0.f;

  for (int c = 0; c < NCH; ++c) {
    if (c + 1 < NCH) {
      issue(c + 1, (c + 1) & 1);
#if USE_ASYNC
      __builtin_amdgcn_s_wait_asynccnt(SEGS);  // chunk c's 6 loads (oldest) done
#endif
    } else {
#if USE_ASYNC
      __builtin_amdgcn_s_wait_asynccnt(0);
#endif
    }
    __syncthreads();  // whole block's chunk-c data visible in LDS

    const float* row = &sv[c & 1][tid * ROWF];
#pragma unroll
    for (int j = 0; j < CH; ++j) {
      const int vi = c * CH + j;
      const float wx = row[j * 3 + 0];
      const float wy = row[j * 3 + 1];
      const float wz = row[j * 3 + 2];
      if (vi == 0) { pvx = wx; pvy = wy; pvz = wz; continue; }

      const int i = vi - 1;  // edge index
      const float ex = wx - pvx, ey = wy - pvy, ez = wz - pvz;
      pvx = wx; pvy = wy; pvz = wz;

      const float rl = restEdgeL[ebase + i];
      const float th = theta[ebase + i];
      float kx = 0.f, ky = 0.f, kz = 0.f;

      if (i == 0) {
        // u0 = normalize(cross(cross(e0, d), e0)); v0 = normalize(cross(e0, u0))
        float nx, ny, nz;
        cross3(ex, ey, ez, dx, dy, dz, nx, ny, nz);
        cross3(nx, ny, nz, ex, ey, ez, ux, uy, uz);
        float inv = 1.f / fmaxf(sqrtf(ux * ux + uy * uy + uz * uz), EPSN);
        ux *= inv; uy *= inv; uz *= inv;
        cross3(ex, ey, ez, ux, uy, uz, vx, vy, vz);
        inv = 1.f / fmaxf(sqrtf(vx * vx + vy * vy + vz * vz), EPSN);
        vx *= inv; vy *= inv; vz *= inv;
      } else {
        // kb = 2*cross(e_{i-1}, e_i) / (rl_{i-1}*rl_i + dot(e_{i-1}, e_i))
        float nx, ny, nz;
        cross3(pex, pey, pez, ex, ey, ez, nx, ny, nz);
        const float den = prl * rl + (pex * ex + pey * ey + pez * ez);
        const float invden = 2.f / den;
        kx = nx * invden; ky = ny * invden; kz = nz * invden;

        const float mag = kx * kx + ky * ky + kz * kz;
        const float invm4 = 1.f / (mag + 4.f);
        const float sinP = sqrtf(mag * invm4);
        const float cosP = sqrtf(4.f * invm4);
        const float invn = 1.f / fmaxf(sqrtf(mag), EPSN);
        const float qvx = sinP * kx * invn;
        const float qvy = sinP * ky * invn;
        const float qvz = sinP * kz * invn;

        // rotate u: t = 2*cross(qv,u); ru = u + cosP*t + cross(qv,t)
        float tx, ty, tz;
        cross3(qvx, qvy, qvz, ux, uy, uz, tx, ty, tz);
        tx *= 2.f; ty *= 2.f; tz *= 2.f;
        float c2x, c2y, c2z;
        cross3(qvx, qvy, qvz, tx, ty, tz, c2x, c2y, c2z);
        const float rux = ux + cosP * tx + c2x;
        const float ruy = uy + cosP * ty + c2y;
        const float ruz = uz + cosP * tz + c2z;
        float rvx, rvy, rvz;
        cross3(ex, ey, ez, rux, ruy, ruz, rvx, rvy, rvz);
        const float invrv =
            1.f / fmaxf(sqrtf(rvx * rvx + rvy * rvy + rvz * rvz), EPSN);
        rvx *= invrv; rvy *= invrv; rvz *= invrv;

        const bool keep = (1.f - cosP) <= ERRC;
        ux = keep ? ux : rux; uy = keep ? uy : ruy; uz = keep ? uz : ruz;
        vx = keep ? vx : rvx; vy = keep ? vy : rvy; vz = keep ? vz : rvz;
      }

      // material frame: m1 = c*u + s*v ; m2 = -s*m1 + c*v  (reference uses new m1)
      float sq, cq;
      __sincosf(th, &sq, &cq);
      const float m1x = cq * ux + sq * vx;
      const float m1y = cq * uy + sq * vy;
      const float m1z = cq * uz + sq * vz;
      const float m2x = -sq * m1x + cq * vx;
      const float m2y = -sq * m1y + cq * vy;
      const float m2z = -sq * m1z + cq * vz;

      const size_t o3 = (ebase + i) * 3;
      om1[o3 + 0] = m1x; om1[o3 + 1] = m1y; om1[o3 + 2] = m1z;
      om2[o3 + 0] = m2x; om2[o3 + 1] = m2y; om2[o3 + 2] = m2z;
      okb[o3 + 0] = kx;  okb[o3 + 1] = ky;  okb[o3 + 2] = kz;

      if (i >= 1) {
        const float owx =  (kx * m2x + ky * m2y + kz * m2z);
        const float owy = -(kx * m1x + ky * m1y + kz * m1z);
        const size_t o2 = (ebase + i) * 2;
        const float dpx = owx - wprev[o2 + 0];
        const float dpy = owy - wprev[o2 + 1];
        const float dnx = owx - wnext[o2 + 0];
        const float dny = owy - wnext[o2 + 1];
        const float bsn = fmaxf(bendS[i], STIFF_THR);   // uniform across lanes
        const float tws = fmaxf(twistS[i], STIFF_THR);  // uniform across lanes
        const float invr = 1.f / restRegionL[ebase + i];
        const float dth = th - pth;
        energy += 0.5f * invr * (bs_last * (dpx * dpx + dpy * dpy) +
                                 bsn * (dnx * dnx + dny * dny) +
                                 tws * dth * dth);
      }

      pex = ex; pey = ey; pez = ez;
      prl = rl; pth = th;
    }
    __syncthreads();  // everyone done reading buf (c&1) before chunk c+2 overwrites it
  }

  oen[b] = energy;
}

extern "C" void kernel_launch(void* const* d_in, const int* in_sizes, int n_in,
                              void* d_out, int out_size, void* d_ws, size_t ws_size,
                              hipStream_t stream) {
  (void)in_sizes; (void)n_in; (void)out_size; (void)d_ws; (void)ws_size;
  const float* vert  = (const float*)d_in[0];
  const float* initd = (const float*)d_in[1];
  const float* rel   = (const float*)d_in[2];
  const float* rrl   = (const float*)d_in[3];
  const float* wp    = (const float*)d_in[4];
  const float* wn    = (const float*)d_in[5];
  const float* th    = (const float*)d_in[6];
  const float* bs    = (const float*)d_in[7];
  const float* ts    = (const float*)d_in[8];
  float* out = (float*)d_out;
  const size_t S = (size_t)B_RODS * NE * 3;  // per 3-vector output block
  der_rod_kernel<<<dim3(B_RODS / RPB), dim3(RPB), 0, stream>>>(
      vert, initd, rel, rrl, wp, wn, th, bs, ts,
      out, out + S, out + 2 * S, out + 3 * S);
}